// GCN2Conv_40690520162783
// MI455X (gfx1250) — compile-verified
//
#include <hip/hip_runtime.h>

typedef __attribute__((ext_vector_type(2))) float v2f;
typedef __attribute__((ext_vector_type(8))) float v8f;

#define NN 100000
#define DD 64
#define EE 1600000
#define ALPHA_C 0.1f
#define BETA_C 0.69314718055994531f

// LDS strides (padded to avoid bank conflicts: bank = (4*row + col) pattern)
#define HSTRIDE 68
#define WSTRIDE 68

// ---------------- zero agg (== d_out) and degree ----------------
__global__ void k_zero(float* __restrict__ out, int* __restrict__ deg) {
    int i = blockIdx.x * blockDim.x + threadIdx.x;
    const int total4 = NN * DD / 4;
    if (i < total4) {
        ((float4*)out)[i] = make_float4(0.f, 0.f, 0.f, 0.f);
    }
    if (i < NN) deg[i] = 0;
}

// ---------------- in-degree ----------------
__global__ void k_degree(const int* __restrict__ dst, int* __restrict__ deg) {
    int e = blockIdx.x * blockDim.x + threadIdx.x;
    if (e < EE) atomicAdd(&deg[dst[e]], 1);
}

// ---------------- norm = clip(deg,1)^-0.5 ----------------
__global__ void k_norm(const int* __restrict__ deg, float* __restrict__ norm) {
    int n = blockIdx.x * blockDim.x + threadIdx.x;
    if (n < NN) {
        float d = (float)deg[n];
        norm[n] = rsqrtf(fmaxf(d, 1.0f));
    }
}

// ---------------- edge scatter: agg[dst] += feat[src]*norm[src] ----------------
// one wave32 per edge; each lane handles 2 of the 64 columns (coalesced 256B/edge)
__global__ __launch_bounds__(256) void k_scatter(const float* __restrict__ feat,
                                                 const float* __restrict__ norm,
                                                 const int* __restrict__ src,
                                                 const int* __restrict__ dst,
                                                 float* __restrict__ agg) {
    int lane = threadIdx.x & 31;
    int wave = (blockIdx.x * blockDim.x + threadIdx.x) >> 5;
    int nWaves = (gridDim.x * blockDim.x) >> 5;
    for (int e = wave; e < EE; e += nWaves) {
        int s = src[e];
        int d = dst[e];
        float ns = norm[s];
        const float2* fs = (const float2*)(feat + (size_t)s * DD);
        float2 v = fs[lane];
        float* ap = agg + (size_t)d * DD + lane * 2;
        unsafeAtomicAdd(ap + 0, v.x * ns);
        unsafeAtomicAdd(ap + 1, v.y * ns);
    }
}

// ---------------- fused combine + GEMM (WMMA f32 16x16x4) + epilogue ----------------
// per wave: one 16-row tile. h = (1-a)*agg*norm + a*feat0 built in LDS,
// then out = (1-b)*h + b*(h @ W) + bias, written in place over d_out.
__global__ __launch_bounds__(256) void k_finalize(float* __restrict__ inout,
                                                  const float* __restrict__ feat0,
                                                  const float* __restrict__ W,
                                                  const float* __restrict__ bias,
                                                  const float* __restrict__ norm) {
    __shared__ float sWt[DD * WSTRIDE];        // W transposed: [n][k], padded
    __shared__ float sH[8][16 * HSTRIDE];      // per-wave 16x64 h tile, padded

    int tid = threadIdx.x;
    int lane = tid & 31;
    int wid = tid >> 5;
    int lo = lane & 15;
    int hi = lane >> 4;

    // cooperative load of W (64x64) into LDS, transposed: sWt[n][k] = W[k][n]
    for (int i = tid; i < DD * DD; i += 256) {
        int k = i >> 6;
        int n = i & 63;
        sWt[n * WSTRIDE + k] = W[i];
    }

    int tile = blockIdx.x * 8 + wid;
    bool valid = tile < (NN / 16);
    int base = tile * 16;

    if (valid) {
        float* hT = &sH[wid][0];
        // build h tile: 16 rows x 64 cols = 256 float4s over 32 lanes
        for (int it = 0; it < 8; ++it) {
            int idx = it * 32 + lane;      // 0..255
            int row = idx >> 4;            // 0..15
            int q = idx & 15;              // float4 index within row
            float nrm = norm[base + row];
            float4 a4 = ((const float4*)(inout + (size_t)(base + row) * DD))[q];
            float4 f4 = ((const float4*)(feat0 + (size_t)(base + row) * DD))[q];
            float4 h4;
            h4.x = (1.f - ALPHA_C) * a4.x * nrm + ALPHA_C * f4.x;
            h4.y = (1.f - ALPHA_C) * a4.y * nrm + ALPHA_C * f4.y;
            h4.z = (1.f - ALPHA_C) * a4.z * nrm + ALPHA_C * f4.z;
            h4.w = (1.f - ALPHA_C) * a4.w * nrm + ALPHA_C * f4.w;
            *(float4*)&hT[row * HSTRIDE + q * 4] = h4;
        }
    }
    __syncthreads();

    if (valid) {
        const float* hT = &sH[wid][0];

        // preload all A fragments: a[k0] = {h[lo][4k0+2hi], h[lo][4k0+2hi+1]}
        v2f a[16];
#pragma unroll
        for (int k0 = 0; k0 < 16; ++k0) {
            a[k0] = *(const v2f*)&hT[lo * HSTRIDE + 4 * k0 + 2 * hi];
        }

#pragma unroll
        for (int n0 = 0; n0 < 4; ++n0) {
            v8f acc = {};
            int n = n0 * 16 + lo;
#pragma unroll
            for (int k0 = 0; k0 < 16; ++k0) {
                // B fragment: {W[4k0+2hi][n], W[4k0+2hi+1][n]} from transposed LDS
                v2f b = *(const v2f*)&sWt[n * WSTRIDE + 4 * k0 + 2 * hi];
                acc = __builtin_amdgcn_wmma_f32_16x16x4_f32(
                    false, a[k0], false, b, (short)0, acc, false, false);
            }
            float bn = bias[n];
#pragma unroll
            for (int r = 0; r < 8; ++r) {
                int m = r + 8 * hi;
                float h = hT[m * HSTRIDE + n];
                inout[(size_t)(base + m) * DD + n] =
                    (1.f - BETA_C) * h + BETA_C * acc[r] + bn;
            }
        }
    }
}

extern "C" void kernel_launch(void* const* d_in, const int* in_sizes, int n_in,
                              void* d_out, int out_size, void* d_ws, size_t ws_size,
                              hipStream_t stream) {
    const float* feat  = (const float*)d_in[0];
    const float* feat0 = (const float*)d_in[1];
    const float* W     = (const float*)d_in[2];
    const float* bias  = (const float*)d_in[3];
    const int*   src   = (const int*)d_in[4];
    const int*   dst   = (const int*)d_in[5];
    float* out = (float*)d_out;

    int* deg = (int*)d_ws;
    float* norm = (float*)((char*)d_ws + (size_t)NN * sizeof(int));

    const int total4 = NN * DD / 4;                       // 1.6M float4s
    k_zero<<<(total4 + 255) / 256, 256, 0, stream>>>(out, deg);
    k_degree<<<(EE + 255) / 256, 256, 0, stream>>>(dst, deg);
    k_norm<<<(NN + 255) / 256, 256, 0, stream>>>(deg, norm);

    // one wave per edge: EE/8 blocks of 8 waves
    k_scatter<<<(EE + 7) / 8, 256, 0, stream>>>(feat, norm, src, dst, out);

    // 6250 row tiles, 8 tiles (waves) per block
    k_finalize<<<(NN / 16 + 7) / 8, 256, 0, stream>>>(out, feat0, W, bias, norm);
}